// MVN_DDI_60808146977242
// MI455X (gfx1250) — compile-verified
//
#include <hip/hip_runtime.h>
#include <math.h>

// ---------------- problem constants (match reference) ----------------
#define BG    2000      // graphs per side
#define NPG   50        // nodes per graph
#define EPG   200       // edges per graph
#define NTOT  (BG*NPG)  // 100000
#define ETOT  (BG*EPG)  // 400000
#define FNODE 70
#define FEDGE 6
#define HID   128
#define EFEAT 64

typedef __attribute__((ext_vector_type(2))) float v2f;
typedef __attribute__((ext_vector_type(4))) float v4f;
typedef __attribute__((ext_vector_type(8))) float v8f;

__device__ __forceinline__ float eluf(float x) { return x > 0.f ? x : expm1f(x); }

// order-preserving float<->uint encoding for atomicMax-based segment max
__device__ __forceinline__ unsigned fenc(float f) {
  unsigned u = __float_as_uint(f);
  return (u & 0x80000000u) ? ~u : (u | 0x80000000u);
}
__device__ __forceinline__ float fdec(unsigned u) {
  unsigned b = (u & 0x80000000u) ? (u & 0x7FFFFFFFu) : ~u;
  return __uint_as_float(b);
}

// ---------------------------------------------------------------------
// GEMM  Y[M x (NT*16)] = act( X[M x K] @ W[K x NT*16] (+ bias) )
// fp32 WMMA 16x16x4, wave32. One wave owns TWO 16-row slabs (shared B).
// W staged in LDS quad-major ([k/4][col][k%4]), zero-padded to 4-row
// multiple. Main loop covers only full in-bounds quads -> affine A
// addresses (imm offsets), no predication; the single ragged quad
// (K=70, K=6) is peeled and address-clamped against the zero padding.
// Per 4-K step: 2x global_load_b64 + NT/2 x ds_load_2addr_b64 + 2*NT WMMA.
// ---------------------------------------------------------------------
template <int NT, bool BIAS, int ACT>
__global__ __launch_bounds__(256) void gemm_wmma_k(
    const float* __restrict__ X, const float* __restrict__ W,
    const float* __restrict__ bias, float* __restrict__ Y, int M, int K) {
  constexpr int NO = NT * 16;
  constexpr int L2NO = (NO == 128) ? 7 : 6;
  __shared__ float sW[128 * NO];  // quad-packed, zero-padded (64KB @ NT=8)

  const int K4p = (K + 3) & ~3;
  for (int i = threadIdx.x; i < K4p * NO; i += 256) {
    const int row = i >> L2NO, col = i & (NO - 1);
    const float v = (row < K) ? W[row * NO + col] : 0.f;
    sW[((((row >> 2) << L2NO) + col) << 2) + (row & 3)] = v;
  }
  __syncthreads();

  const int wave = threadIdx.x >> 5, lane = threadIdx.x & 31;
  const int mrow = lane & 15;
  const int khalf = (lane >> 4) << 1;  // 0 or 2
  const int row0 = (blockIdx.x * 8 + wave) * 32;

  int r0 = row0 + mrow;      if (r0 > M - 1) r0 = M - 1;
  int r1 = row0 + 16 + mrow; if (r1 > M - 1) r1 = M - 1;
  const float* x0 = X + (size_t)r0 * K + khalf;
  const float* x1 = X + (size_t)r1 * K + khalf;

  v8f acc0[NT] = {}, acc1[NT] = {};
  const float* wq = sW + (mrow << 2) + khalf;  // lane base within a quad row

  const int Kmain = K & ~3;  // full quads: kk+1 <= k+3 <= K-1, no guards
  for (int k = 0; k < Kmain; k += 4) {
    const v2f a0 = *(const v2f*)(x0 + k);
    const v2f a1 = *(const v2f*)(x1 + k);
    const float* wk = wq + ((k >> 2) << (L2NO + 2));
    v2f bf[NT];
#pragma unroll
    for (int t = 0; t < NT; ++t) bf[t] = *(const v2f*)(wk + (t << 6));
#pragma unroll
    for (int t = 0; t < NT; ++t) {
      acc0[t] = __builtin_amdgcn_wmma_f32_16x16x4_f32(
          false, a0, false, bf[t], (short)0, acc0[t], false, false);
      acc1[t] = __builtin_amdgcn_wmma_f32_16x16x4_f32(
          false, a1, false, bf[t], (short)0, acc1[t], false, false);
    }
  }
  if (Kmain < K4p) {  // peeled ragged quad (K % 4 != 0): clamp A, B rows
    const int kk = Kmain + khalf;                  //   beyond K are zeros
    const int kc = (kk + 1 < K) ? kk : (K - 2);    //   so clamp garbage *0
    const v2f a0 = *(const v2f*)(x0 - khalf + kc);
    const v2f a1 = *(const v2f*)(x1 - khalf + kc);
    const float* wk = wq + ((Kmain >> 2) << (L2NO + 2));
    v2f bf[NT];
#pragma unroll
    for (int t = 0; t < NT; ++t) bf[t] = *(const v2f*)(wk + (t << 6));
#pragma unroll
    for (int t = 0; t < NT; ++t) {
      acc0[t] = __builtin_amdgcn_wmma_f32_16x16x4_f32(
          false, a0, false, bf[t], (short)0, acc0[t], false, false);
      acc1[t] = __builtin_amdgcn_wmma_f32_16x16x4_f32(
          false, a1, false, bf[t], (short)0, acc1[t], false, false);
    }
  }

  const int cloc = lane & 15;
  const int roff = (lane >> 4) << 3;  // +8 for upper half-lanes
#pragma unroll
  for (int t = 0; t < NT; ++t) {
    const int col = t * 16 + cloc;
    const float bv = BIAS ? bias[col] : 0.f;
#pragma unroll
    for (int j = 0; j < 8; ++j) {
      int rr = row0 + roff + j;
      if (rr < M) {
        float yv = acc0[t][j] + bv;
        if (ACT == 1) yv = eluf(yv);
        Y[(size_t)rr * NO + col] = yv;
      }
      rr += 16;
      if (rr < M) {
        float yv = acc1[t][j] + bv;
        if (ACT == 1) yv = eluf(yv);
        Y[(size_t)rr * NO + col] = yv;
      }
    }
  }
}

// ------------------- graph LayerNorm (mode='graph') + optional ELU ----
__global__ __launch_bounds__(256) void gln_k(const float* __restrict__ in,
                                             float* __restrict__ out,
                                             const float* __restrict__ gam,
                                             const float* __restrict__ bet,
                                             int act) {
  __shared__ float rs_[8], r2_[8], mv[2];
  const int g = blockIdx.x;
  const float* xg = in + (size_t)g * NPG * HID;
  float s = 0.f, s2 = 0.f;
  for (int i = threadIdx.x; i < NPG * HID; i += 256) {
    float v = xg[i]; s += v; s2 += v * v;
  }
  for (int m = 16; m >= 1; m >>= 1) {
    s += __shfl_xor(s, m, 32); s2 += __shfl_xor(s2, m, 32);
  }
  const int wave = threadIdx.x >> 5, lane = threadIdx.x & 31;
  if (lane == 0) { rs_[wave] = s; r2_[wave] = s2; }
  __syncthreads();
  if (threadIdx.x == 0) {
    float S = 0.f, S2 = 0.f;
    for (int w = 0; w < 8; ++w) { S += rs_[w]; S2 += r2_[w]; }
    const float cnt = (float)(NPG * HID);
    float mean = S / cnt;
    float var = S2 / cnt - mean * mean;
    mv[0] = mean; mv[1] = rsqrtf(var + 1e-5f);
  }
  __syncthreads();
  const float mean = mv[0], rs = mv[1];
  float* yg = out + (size_t)g * NPG * HID;
  for (int i = threadIdx.x; i < NPG * HID; i += 256) {
    const int c = i & (HID - 1);
    float y = (xg[i] - mean) * rs * gam[c] + bet[c];
    if (act) y = eluf(y);
    yg[i] = y;
  }
}

// ------------- attention: logits + per-dst running max (wave/edge) ----
__global__ __launch_bounds__(256) void alpha_k(const float* __restrict__ q,
                                               const float* __restrict__ k,
                                               const float* __restrict__ ep,
                                               const int* __restrict__ src,
                                               const int* __restrict__ dst,
                                               float* __restrict__ exb,
                                               unsigned* __restrict__ mx,
                                               int ecount) {
  const int e = blockIdx.x * 8 + (threadIdx.x >> 5);
  if (e >= ecount) return;
  const int lane = threadIdx.x & 31;
  const int s = src[e], d = dst[e];
  const int c0 = lane * 4;
  const v4f qv = *(const v4f*)(q + (size_t)d * HID + c0);
  const v4f kv = *(const v4f*)(k + (size_t)s * HID + c0);
  const v4f ev = *(const v4f*)(ep + (size_t)e * HID + c0);
  float p = 0.f;
#pragma unroll
  for (int j = 0; j < 4; ++j) p += qv[j] * (kv[j] + ev[j]);
  p += __shfl_xor(p, 1, 32); p += __shfl_xor(p, 2, 32);
  p += __shfl_xor(p, 4, 32); p += __shfl_xor(p, 8, 32);
  p *= 0.125f;  // / sqrt(DHEAD=64)
  if ((lane & 15) == 0) {
    const int head = lane >> 4;
    exb[(size_t)e * 2 + head] = p;
    atomicMax(&mx[(size_t)d * 2 + head], fenc(p));
  }
}

__global__ __launch_bounds__(256) void zero2_k(unsigned* __restrict__ mx,
                                               float* __restrict__ den, int n) {
  const int i = blockIdx.x * 256 + threadIdx.x;
  if (i < n) { mx[i] = 0u; den[i] = 0.f; }
}

__global__ __launch_bounds__(256) void expden_k(float* __restrict__ exb,
                                                const unsigned* __restrict__ mx,
                                                float* __restrict__ den,
                                                const int* __restrict__ dst,
                                                int ecount) {
  const int i = blockIdx.x * 256 + threadIdx.x;
  if (i >= ecount * 2) return;
  const int e = i >> 1, h = i & 1;
  const int d = dst[e];
  const float m = fdec(mx[d * 2 + h]);
  const float v = expf(exb[i] - m);
  exb[i] = v;
  atomicAdd(&den[d * 2 + h], v);
}

// ---- message aggregation per graph: out = segsum((v[src]+e)*alpha)+skip
__global__ __launch_bounds__(256) void msg_agg_k(const float* __restrict__ v,
                                                 const float* __restrict__ ep,
                                                 const float* __restrict__ sk,
                                                 const float* __restrict__ exb,
                                                 const float* __restrict__ den,
                                                 const int* __restrict__ src,
                                                 const int* __restrict__ dst,
                                                 float* __restrict__ out) {
  __shared__ float sacc[NPG * HID];
  const int g = blockIdx.x;
  for (int i = threadIdx.x; i < NPG * HID; i += 256) sacc[i] = 0.f;
  __syncthreads();
  const int wave = threadIdx.x >> 5, lane = threadIdx.x & 31;
  const int c0 = lane * 4, head = lane >> 4;
  for (int el = wave; el < EPG; el += 8) {
    const int e = g * EPG + el;
    const int s = src[e], d = dst[e];
    const float a = exb[(size_t)e * 2 + head] / (den[(size_t)d * 2 + head] + 1e-16f);
    const int dl = d - g * NPG;
    const v4f vv = *(const v4f*)(v + (size_t)s * HID + c0);
    const v4f ev = *(const v4f*)(ep + (size_t)e * HID + c0);
    float* ap = sacc + dl * HID + c0;
#pragma unroll
    for (int j = 0; j < 4; ++j) atomicAdd(&ap[j], (vv[j] + ev[j]) * a);
  }
  __syncthreads();
  const size_t base = (size_t)g * NPG * HID;
  for (int i = threadIdx.x; i < NPG * HID; i += 256)
    out[base + i] = sacc[i] + sk[base + i];
}

// ---------------- GCN linear: xw[i] = dot(x[i,:HID], W[:,0]) ----------
__global__ __launch_bounds__(256) void gcnlin_k(const float* __restrict__ x,
                                                const float* __restrict__ W,
                                                float* __restrict__ xw, int n) {
  const int wid = blockIdx.x * 8 + (threadIdx.x >> 5);
  if (wid >= n) return;
  const int lane = threadIdx.x & 31;
  const v4f xr = *(const v4f*)(x + (size_t)wid * HID + lane * 4);
  const v4f wr = *(const v4f*)(W + lane * 4);
  float s = xr[0] * wr[0] + xr[1] * wr[1] + xr[2] * wr[2] + xr[3] * wr[3];
  for (int m = 16; m >= 1; m >>= 1) s += __shfl_xor(s, m, 32);
  if (lane == 0) xw[wid] = s;
}

// --- per-graph: GCN score softmax, SAG pool, edge pool, emb + l2norm --
__global__ __launch_bounds__(256) void pool_k(const float* __restrict__ x2,
                                              const float* __restrict__ xw,
                                              const float* __restrict__ ea2,
                                              const int* __restrict__ src,
                                              const int* __restrict__ dst,
                                              const float* __restrict__ gcnb,
                                              const float* __restrict__ reW,
                                              const float* __restrict__ reb,
                                              float* __restrict__ emb) {
  __shared__ int scnt[NPG];
  __shared__ float sxw[NPG], sdinv[NPG], sagg[NPG], ssc[NPG];
  __shared__ float sep[EFEAT], semb[HID], red2[2];
  const int g = blockIdx.x, tid = threadIdx.x;
  if (tid < NPG) { scnt[tid] = 0; sagg[tid] = 0.f; }
  __syncthreads();
  for (int el = tid; el < EPG; el += 256)
    atomicAdd(&scnt[dst[g * EPG + el] - g * NPG], 1);
  if (tid < EFEAT) {  // edge pooling (pre-ELU ea after lin_up2)
    float s = 0.f;
    const float* ep0 = ea2 + (size_t)g * EPG * EFEAT + tid;
    for (int el = 0; el < EPG; ++el) s += ep0[(size_t)el * EFEAT];
    sep[tid] = s;
  }
  __syncthreads();
  if (tid < NPG) {
    sdinv[tid] = rsqrtf((float)(scnt[tid] + 1));
    sxw[tid] = xw[g * NPG + tid];
  }
  __syncthreads();
  for (int el = tid; el < EPG; el += 256) {
    const int e = g * EPG + el;
    const int sl = src[e] - g * NPG, dl = dst[e] - g * NPG;
    atomicAdd(&sagg[dl], sxw[sl] * sdinv[sl] * sdinv[dl]);
  }
  __syncthreads();
  if (tid < NPG)
    ssc[tid] = sagg[tid] + sxw[tid] / (float)(scnt[tid] + 1) + gcnb[0];
  __syncthreads();
  if (tid == 0) {
    float m = -3.4e38f;
    for (int i = 0; i < NPG; ++i) m = fmaxf(m, ssc[i]);
    float dn = 0.f;
    for (int i = 0; i < NPG; ++i) dn += expf(ssc[i] - m);
    red2[0] = m; red2[1] = dn;
  }
  __syncthreads();
  if (tid < NPG) ssc[tid] = expf(ssc[tid] - red2[0]) / (red2[1] + 1e-16f);
  __syncthreads();
  if (tid < HID) {
    float pool = 0.f;
    const float* xg = x2 + (size_t)g * NPG * HID + tid;
    for (int i = 0; i < NPG; ++i) pool += xg[(size_t)i * HID] * ssc[i];
    float er = reb[tid];
    for (int k2 = 0; k2 < EFEAT; ++k2) er += sep[k2] * reW[k2 * HID + tid];
    semb[tid] = pool * eluf(er);
  }
  __syncthreads();
  if (tid == 0) {
    float s = 0.f;
    for (int i = 0; i < HID; ++i) s += semb[i] * semb[i];
    red2[0] = fmaxf(sqrtf(s), 1e-12f);
  }
  __syncthreads();
  if (tid < HID) emb[(size_t)g * HID + tid] = eluf(semb[tid] / red2[0]);
}

// -------------------------- elementwise ELU ---------------------------
__global__ __launch_bounds__(256) void elu_k(float* __restrict__ x, int n) {
  const int i = blockIdx.x * 256 + threadIdx.x;
  if (i < n) x[i] = eluf(x[i]);
}

// --------------------- 0.6*embA + 0.4*embB -> out ---------------------
__global__ __launch_bounds__(256) void combine_k(const float* __restrict__ a,
                                                 const float* __restrict__ b,
                                                 float* __restrict__ out, int n) {
  const int i = blockIdx.x * 256 + threadIdx.x;
  if (i < n) out[i] = 0.6f * a[i] + 0.4f * b[i];
}

// ======================================================================
// Host-side orchestration
// ======================================================================
struct ConvP { const float *eW, *kW, *kb, *qW, *qb, *skW, *skb, *vW, *vb; };
struct BlockP {
  ConvP c1, c2;
  const float *gcnW, *gcnb, *upW, *upb, *up2W, *up2b;
  const float *n_b, *n_g, *n2_b, *n2_g, *reW, *reb;
};
struct Params {
  BlockP blk[2];
  const float *e0W, *e0b, *n0W, *n0b, *nm0_b, *nm0_g;
};
struct Work {
  float *x, *t0, *q, *k, *v, *sk, *ep, *eaA, *eaB, *exb, *den, *xw, *embA, *embB;
  unsigned* mx;
};

static inline int dgrid(int n, int b) { return (n + b - 1) / b; }

static void run_conv(const ConvP& c, const float* x, const float* ea,
                     const int* src, const int* dst, const Work& w,
                     float* out, hipStream_t st) {
  // projections (fp32 WMMA); 256 rows per block (8 waves x 2 slabs x 16)
  gemm_wmma_k<8, true, 0><<<dgrid(NTOT, 256), 256, 0, st>>>(x, c.qW, c.qb, w.q, NTOT, HID);
  gemm_wmma_k<8, true, 0><<<dgrid(NTOT, 256), 256, 0, st>>>(x, c.kW, c.kb, w.k, NTOT, HID);
  gemm_wmma_k<8, true, 0><<<dgrid(NTOT, 256), 256, 0, st>>>(x, c.vW, c.vb, w.v, NTOT, HID);
  gemm_wmma_k<8, true, 0><<<dgrid(NTOT, 256), 256, 0, st>>>(x, c.skW, c.skb, w.sk, NTOT, HID);
  gemm_wmma_k<8, false, 0><<<dgrid(ETOT, 256), 256, 0, st>>>(ea, c.eW, nullptr, w.ep, ETOT, EFEAT);
  // segment softmax over incoming edges per destination
  zero2_k<<<dgrid(NTOT * 2, 256), 256, 0, st>>>(w.mx, w.den, NTOT * 2);
  alpha_k<<<dgrid(ETOT, 8), 256, 0, st>>>(w.q, w.k, w.ep, src, dst, w.exb, w.mx, ETOT);
  expden_k<<<dgrid(ETOT * 2, 256), 256, 0, st>>>(w.exb, w.mx, w.den, dst, ETOT);
  // aggregate messages + skip
  msg_agg_k<<<BG, 256, 0, st>>>(w.v, w.ep, w.sk, w.exb, w.den, src, dst, out);
}

static void run_side(const float* xin, const int* ei, const float* eain,
                     const Params& P, const Work& w, float* out_side,
                     hipStream_t st) {
  const int* src = ei;
  const int* dst = ei + ETOT;
  // node embed -> graph LN -> ELU
  gemm_wmma_k<8, true, 0><<<dgrid(NTOT, 256), 256, 0, st>>>(xin, P.n0W, P.n0b, w.t0, NTOT, FNODE);
  gln_k<<<BG, 256, 0, st>>>(w.t0, w.x, P.nm0_g, P.nm0_b, 1);
  // edge embed -> ELU
  gemm_wmma_k<4, true, 1><<<dgrid(ETOT, 256), 256, 0, st>>>(eain, P.e0W, P.e0b, w.eaA, ETOT, FEDGE);

  float* embs[2] = {w.embA, w.embB};
  for (int b = 0; b < 2; ++b) {
    const BlockP& bp = P.blk[b];
    // conv1 -> graph LN -> ELU
    run_conv(bp.c1, w.x, w.eaA, src, dst, w, w.t0, st);
    gln_k<<<BG, 256, 0, st>>>(w.t0, w.x, bp.n_g, bp.n_b, 1);
    // lin_up (ELU fused)
    gemm_wmma_k<4, true, 1><<<dgrid(ETOT, 256), 256, 0, st>>>(w.eaA, bp.upW, bp.upb, w.eaB, ETOT, EFEAT);
    // conv2 (output w.t0 = x2, pre-norm2)
    run_conv(bp.c2, w.x, w.eaB, src, dst, w, w.t0, st);
    // lin_up2 (no act) -> eaA holds ea2 (pre-ELU)
    gemm_wmma_k<4, true, 0><<<dgrid(ETOT, 256), 256, 0, st>>>(w.eaB, bp.up2W, bp.up2b, w.eaA, ETOT, EFEAT);
    // GCN score + softmax + SAG pool + edge pool + emb + l2norm + ELU
    gcnlin_k<<<dgrid(NTOT, 8), 256, 0, st>>>(w.t0, bp.gcnW, w.xw, NTOT);
    pool_k<<<BG, 256, 0, st>>>(w.t0, w.xw, w.eaA, src, dst, bp.gcnb, bp.reW, bp.reb, embs[b]);
    // x = elu(graph_ln(x2, norm2)); ea = elu(ea2)
    gln_k<<<BG, 256, 0, st>>>(w.t0, w.x, bp.n2_g, bp.n2_b, 1);
    elu_k<<<dgrid(ETOT * EFEAT, 256), 256, 0, st>>>(w.eaA, ETOT * EFEAT);
  }
  combine_k<<<dgrid(BG * HID, 256), 256, 0, st>>>(w.embA, w.embB, out_side, BG * HID);
}

extern "C" void kernel_launch(void* const* d_in, const int* in_sizes, int n_in,
                              void* d_out, int out_size, void* d_ws, size_t ws_size,
                              hipStream_t stream) {
  (void)in_sizes; (void)n_in; (void)out_size; (void)ws_size;
  // ---- data inputs (setup_inputs dict order) ----
  const float* h_x  = (const float*)d_in[0];
  const int*   h_ei = (const int*)d_in[1];
  const float* h_ea = (const float*)d_in[2];
  const float* t_x  = (const float*)d_in[5];
  const int*   t_ei = (const int*)d_in[6];
  const float* t_ea = (const float*)d_in[7];

  // ---- params: JAX pytree leaf order (alphabetical dict keys) ----
  int ix = 10;
  auto F = [&]() { return (const float*)d_in[ix++]; };
  Params P;
  for (int b = 0; b < 2; ++b) {
    ConvP* cs[2] = {&P.blk[b].c1, &P.blk[b].c2};
    for (int c = 0; c < 2; ++c) {
      ConvP& cp = *cs[c];
      cp.eW = F();                // e: {W}
      cp.kW = F(); cp.kb = F();   // k: {W,b}
      cp.qW = F(); cp.qb = F();   // q
      cp.skW = F(); cp.skb = F(); // skip
      cp.vW = F(); cp.vb = F();   // v
    }
    P.blk[b].gcnW = F(); P.blk[b].gcnb = F();
    P.blk[b].upW = F();  P.blk[b].upb = F();
    P.blk[b].up2W = F(); P.blk[b].up2b = F();
    P.blk[b].n_b = F();  P.blk[b].n_g = F();   // norm: {b,g}
    P.blk[b].n2_b = F(); P.blk[b].n2_g = F();  // norm2
    P.blk[b].reW = F();  P.blk[b].reb = F();
  }
  P.e0W = F(); P.e0b = F();
  P.n0W = F(); P.n0b = F();
  P.nm0_b = F(); P.nm0_g = F();

  // ---- workspace carve-up (256B aligned) ----
  char* base = (char*)d_ws;
  size_t off = 0;
  auto alloc = [&](size_t bytes) {
    void* p = base + off;
    off += (bytes + 255) & ~(size_t)255;
    return p;
  };
  Work w;
  w.x   = (float*)alloc((size_t)NTOT * HID * 4);
  w.t0  = (float*)alloc((size_t)NTOT * HID * 4);
  w.q   = (float*)alloc((size_t)NTOT * HID * 4);
  w.k   = (float*)alloc((size_t)NTOT * HID * 4);
  w.v   = (float*)alloc((size_t)NTOT * HID * 4);
  w.sk  = (float*)alloc((size_t)NTOT * HID * 4);
  w.ep  = (float*)alloc((size_t)ETOT * HID * 4);
  w.eaA = (float*)alloc((size_t)ETOT * EFEAT * 4);
  w.eaB = (float*)alloc((size_t)ETOT * EFEAT * 4);
  w.exb = (float*)alloc((size_t)ETOT * 2 * 4);
  w.mx  = (unsigned*)alloc((size_t)NTOT * 2 * 4);
  w.den = (float*)alloc((size_t)NTOT * 2 * 4);
  w.xw  = (float*)alloc((size_t)NTOT * 4);
  w.embA = (float*)alloc((size_t)BG * HID * 4);
  w.embB = (float*)alloc((size_t)BG * HID * 4);

  float* out = (float*)d_out;
  run_side(h_x, h_ei, h_ea, P, w, out, stream);                     // h_fin
  run_side(t_x, t_ei, t_ea, P, w, out + (size_t)BG * HID, stream);  // t_fin
}